// Net_29661044146779
// MI455X (gfx1250) — compile-verified
//
#include <hip/hip_runtime.h>
#include <hip/hip_bf16.h>
#include <cstdint>
#include <cstddef>

#define FD 128            // feature dim D
#define EPS_AGG 1e-6f
#define EPS_BN  1e-5f

typedef __attribute__((ext_vector_type(2))) float v2f;
typedef __attribute__((ext_vector_type(8))) float v8f;

// ---------------------------------------------------------------------------
// ws layout (floats):
//   [0,128)     colsum
//   [128,256)   colsumsq
//   [256,384)   mean
//   [384,512)   scale = rsqrt(var+eps)*gamma
//   [512 ...)   num_f[N*D], den_f[N*D], num_b[N*D], den_b[N*D]
// ---------------------------------------------------------------------------

__global__ void k_zero(float4* __restrict__ p, long n4) {
    long i = (long)blockIdx.x * blockDim.x + threadIdx.x;
    const long stride = (long)gridDim.x * blockDim.x;
    const float4 z = make_float4(0.f, 0.f, 0.f, 0.f);
    for (; i < n4; i += stride) p[i] = z;
}

// One wave = one edge. Lane l handles features [4l, 4l+4): one float4 per lane
// covers the whole 512B row, fully coalesced. 16 hardware f32 atomics per lane
// into L2-resident accumulators.
__global__ void k_edge(const float* __restrict__ h, const float* __restrict__ e,
                       const int* __restrict__ src, const int* __restrict__ dst,
                       float* __restrict__ num_f, float* __restrict__ den_f,
                       float* __restrict__ num_b, float* __restrict__ den_b,
                       int nE) {
    const int lane   = threadIdx.x & 31;
    const int wave   = blockIdx.x * (blockDim.x >> 5) + (threadIdx.x >> 5);
    const int nwaves = gridDim.x * (blockDim.x >> 5);
    const float4* h4 = (const float4*)h;
    const float4* e4 = (const float4*)e;

    for (int eid = wave; eid < nE; eid += nwaves) {
        const int s = src[eid];
        const int t = dst[eid];
        const int nxt = eid + nwaves;
        if (nxt < nE)  // CDNA5 global_prefetch_b8 on the next streamed e row
            __builtin_prefetch((const void*)&e4[(size_t)nxt * 32 + lane], 0, 1);

        const float4 hs = h4[(size_t)s * 32 + lane];
        const float4 ht = h4[(size_t)t * 32 + lane];
        const float4 ev = e4[(size_t)eid * 32 + lane];

        const float hsv[4] = {hs.x, hs.y, hs.z, hs.w};
        const float htv[4] = {ht.x, ht.y, ht.z, ht.w};
        const float evv[4] = {ev.x, ev.y, ev.z, ev.w};

        const size_t tb = (size_t)t * FD + (size_t)lane * 4;
        const size_t sb = (size_t)s * FD + (size_t)lane * 4;
#pragma unroll
        for (int k = 0; k < 4; ++k) {
            const float x = 0.1f * hsv[k] - 10.0f * htv[k] + evv[k];
            const float g = 1.0f / (1.0f + __expf(-x));
            unsafeAtomicAdd(&num_f[tb + k], g * 100.0f * hsv[k]);
            unsafeAtomicAdd(&den_f[tb + k], g);
            unsafeAtomicAdd(&num_b[sb + k], -g * htv[k]);
            unsafeAtomicAdd(&den_b[sb + k], g);
        }
    }
}

// h2 = 10*h + num_f/(den_f+eps) + num_b/(den_b+eps); written into d_out (scratch).
__global__ void k_node(const float4* __restrict__ h, const float4* __restrict__ num_f,
                       const float4* __restrict__ den_f, const float4* __restrict__ num_b,
                       const float4* __restrict__ den_b, float4* __restrict__ h2, long n4) {
    long i = (long)blockIdx.x * blockDim.x + threadIdx.x;
    const long stride = (long)gridDim.x * blockDim.x;
    for (; i < n4; i += stride) {
        const float4 hv = h[i], nf = num_f[i], df = den_f[i], nb = num_b[i], db = den_b[i];
        float4 r;
        r.x = 10.0f * hv.x + nf.x / (df.x + EPS_AGG) + nb.x / (db.x + EPS_AGG);
        r.y = 10.0f * hv.y + nf.y / (df.y + EPS_AGG) + nb.y / (db.y + EPS_AGG);
        r.z = 10.0f * hv.z + nf.z / (df.z + EPS_AGG) + nb.z / (db.z + EPS_AGG);
        r.w = 10.0f * hv.w + nf.w / (df.w + EPS_AGG) + nb.w / (db.w + EPS_AGG);
        h2[i] = r;
    }
}

// Column sums of h2 and h2^2 on the matrix pipe: V_WMMA_F32_16X16X4_F32 with
// A == all-ones(16x4) => C[i][j] = sum_k B[k][j]. One wave reduces a 16-column
// slice over a 512-row block, 4 rows per WMMA; result read from c[0], lane j.
// With A all-ones the result is invariant to which of B's 4 register slots
// holds which row, so the exact B layout cannot affect correctness.
__global__ void k_stats(const float* __restrict__ h2, float* __restrict__ colsum,
                        float* __restrict__ colsumsq, int nRows) {
    const int lane   = threadIdx.x & 31;
    const int wave   = blockIdx.x * (blockDim.x >> 5) + (threadIdx.x >> 5);
    const int nwaves = gridDim.x * (blockDim.x >> 5);
    const int ROWS   = 512;
    const int nRowBlk = (nRows + ROWS - 1) / ROWS;
    const int nSlices = FD / 16;
    const int nWork  = nRowBlk * nSlices;

    v2f a; a.x = 1.0f; a.y = 1.0f;            // all-ones A matrix (16x4)
    const int ln   = lane & 15;
    const int off0 = (lane < 16) ? 0 : 2;

    for (int w = wave; w < nWork; w += nwaves) {
        const int slice = w % nSlices;
        const int rb    = w / nSlices;
        const int cb    = slice * 16;
        const int r0    = rb * ROWS;
        const int r1    = (r0 + ROWS < nRows) ? (r0 + ROWS) : nRows;

        v8f cs = {};
        v8f cq = {};
        for (int r = r0; r + 3 < r1; r += 4) {
            const float b0 = h2[(size_t)(r + off0)     * FD + cb + ln];
            const float b1 = h2[(size_t)(r + off0 + 1) * FD + cb + ln];
            v2f b;  b.x  = b0;      b.y  = b1;
            v2f bq; bq.x = b0 * b0; bq.y = b1 * b1;
            cs = __builtin_amdgcn_wmma_f32_16x16x4_f32(false, a, false, b,
                                                       (short)0, cs, false, false);
            cq = __builtin_amdgcn_wmma_f32_16x16x4_f32(false, a, false, bq,
                                                       (short)0, cq, false, false);
        }
        if (lane < 16) {
            unsafeAtomicAdd(&colsum[cb + lane],   cs[0]);
            unsafeAtomicAdd(&colsumsq[cb + lane], cq[0]);
        }
    }
}

__global__ void k_bnstats(const float* __restrict__ colsum, const float* __restrict__ colsumsq,
                          const float* __restrict__ gamma, float* __restrict__ mean,
                          float* __restrict__ scale, float invN) {
    const int d = threadIdx.x;
    const float m = colsum[d] * invN;
    float v = colsumsq[d] * invN - m * m;   // biased variance, E[x^2]-E[x]^2
    if (v < 0.f) v = 0.f;
    mean[d]  = m;
    scale[d] = rsqrtf(v + EPS_BN) * gamma[d];
}

// out = relu((h2 - mean)*scale + beta) + h_in ; h2 and out share d_out.
__global__ void k_final(float4* __restrict__ io, const float4* __restrict__ h_in,
                        const float* __restrict__ mean, const float* __restrict__ scale,
                        const float* __restrict__ beta, long n4) {
    long i = (long)blockIdx.x * blockDim.x + threadIdx.x;
    const long stride = (long)gridDim.x * blockDim.x;
    for (; i < n4; i += stride) {
        const float4 v = io[i];
        const float4 r = h_in[i];
        const int dbase = (int)(i & (FD / 4 - 1)) * 4;   // 32 float4 per row
        float vv[4] = {v.x, v.y, v.z, v.w};
        const float rr[4] = {r.x, r.y, r.z, r.w};
        float4 o;
        float* ov = &o.x;
#pragma unroll
        for (int k = 0; k < 4; ++k) {
            const int d = dbase + k;
            float x = (vv[k] - mean[d]) * scale[d] + beta[d];
            x = (x > 0.f) ? x : 0.f;
            ov[k] = x + rr[k];
        }
        io[i] = o;
    }
}

extern "C" void kernel_launch(void* const* d_in, const int* in_sizes, int n_in,
                              void* d_out, int out_size, void* d_ws, size_t ws_size,
                              hipStream_t stream) {
    const float* h     = (const float*)d_in[0];
    const float* e     = (const float*)d_in[1];
    const int*   src   = (const int*)d_in[2];
    const int*   dst   = (const int*)d_in[3];
    const float* gamma = (const float*)d_in[4];
    const float* beta  = (const float*)d_in[5];
    float* out = (float*)d_out;

    const int N  = in_sizes[0] / FD;   // 50000
    const int nE = in_sizes[2];        // 600000
    const size_t nd = (size_t)N * FD;

    float* ws       = (float*)d_ws;
    float* colsum   = ws;
    float* colsumsq = ws + FD;
    float* mean     = ws + 2 * FD;
    float* scale    = ws + 3 * FD;
    float* num_f    = ws + 4 * FD;
    float* den_f    = num_f + nd;
    float* num_b    = den_f + nd;
    float* den_b    = num_b + nd;

    // Zero accumulators + stat sums (must happen every call; harness doesn't re-poison).
    const long zero4 = (long)((4 * FD + 4 * nd) / 4);
    k_zero<<<2048, 256, 0, stream>>>((float4*)ws, zero4);

    k_edge<<<4096, 256, 0, stream>>>(h, e, src, dst, num_f, den_f, num_b, den_b, nE);

    const long n4 = (long)(nd / 4);
    k_node<<<2048, 256, 0, stream>>>((const float4*)h, (const float4*)num_f,
                                     (const float4*)den_f, (const float4*)num_b,
                                     (const float4*)den_b, (float4*)out, n4);

    k_stats<<<104, 256, 0, stream>>>(out, colsum, colsumsq, N);

    k_bnstats<<<1, FD, 0, stream>>>(colsum, colsumsq, gamma, mean, scale, 1.0f / (float)N);

    k_final<<<2048, 256, 0, stream>>>((float4*)out, (const float4*)h, mean, scale, beta, n4);
}